// MAGC_16303695855656
// MI455X (gfx1250) — compile-verified
//
#include <hip/hip_runtime.h>

// ---------------- problem constants (from reference) ----------------
#define B_    16
#define H_    48
#define W_    160
#define C_    512
#define G_    8
#define SHI   64              // C_/G_
#define N_    (H_*W_)         // 7680 spatial positions
#define CHUNKS   64           // chunks per (b,g) for the online-softmax pass
#define CHUNK_N  (N_/CHUNKS)  // 120 positions per chunk (exact)
#define PSTRIDE  66           // 64 acc + m + s per chunk partial
#define LN_EPS   1e-3f

typedef __attribute__((ext_vector_type(2))) float v2f;
typedef __attribute__((ext_vector_type(4))) float v4f;
typedef __attribute__((ext_vector_type(8))) float v8f;

// ------------------------------------------------------------------
// Kernel 1: fused logits + online softmax + weighted pooling.
// One wave per (b, g, chunk). Wave reads 64 contiguous floats per
// position (float2/lane), reduces the mask dot across the wave,
// and keeps a flash-attention style running (max, sum, acc[64]).
// Reads x exactly once (252 MB) for the whole attention stage.
// ------------------------------------------------------------------
__global__ void attn_pool_partial(const float* __restrict__ x,
                                  const float* __restrict__ w_mask,
                                  const float* __restrict__ b_mask,
                                  float* __restrict__ partial) {
  const int wave = (int)((blockIdx.x * blockDim.x + threadIdx.x) >> 5);
  const int lane = (int)(threadIdx.x & 31);
  const int chunk = wave & (CHUNKS - 1);
  const int bg = wave >> 6;               // log2(CHUNKS) == 6
  if (bg >= B_ * G_) return;
  const int b = bg >> 3;
  const int g = bg & (G_ - 1);

  const float* base = x + ((size_t)b * N_ + (size_t)chunk * CHUNK_N) * C_
                        + (size_t)g * SHI + (size_t)lane * 2;
  const float wm0 = w_mask[lane * 2 + 0];
  const float wm1 = w_mask[lane * 2 + 1];
  const float bm  = b_mask[0];
  const float rsq = 0.125f;               // 1/sqrt(shi=64)

  float m = -3.0e38f, s = 0.0f;
  float ax = 0.0f, ay = 0.0f;             // lane holds channels 2*lane, 2*lane+1

  for (int i = 0; i < CHUNK_N; ++i) {
    v2f f = *(const v2f*)(base + (size_t)i * C_);
    float d = f.x * wm0 + f.y * wm1;
    // wave32 butterfly reduction -> every lane gets the full 64-ch dot
    d += __shfl_xor(d, 16, 32);
    d += __shfl_xor(d, 8, 32);
    d += __shfl_xor(d, 4, 32);
    d += __shfl_xor(d, 2, 32);
    d += __shfl_xor(d, 1, 32);
    const float logit = (d + bm) * rsq;
    const float mn    = fmaxf(m, logit);
    const float alpha = __expf(m - mn);       // 0 on first iteration
    const float p     = __expf(logit - mn);
    ax = ax * alpha + p * f.x;
    ay = ay * alpha + p * f.y;
    s  = s  * alpha + p;
    m  = mn;
  }

  float* pp = partial + (size_t)wave * PSTRIDE;
  pp[lane * 2 + 0] = ax;
  pp[lane * 2 + 1] = ay;
  if (lane == 0) { pp[64] = m; pp[65] = s; }
}

// ------------------------------------------------------------------
// Kernel 2: merge CHUNKS partials per (b,g) -> ctx[16,512]
// ------------------------------------------------------------------
__global__ void attn_combine(const float* __restrict__ partial,
                             float* __restrict__ ctx) {
  const int bg = (int)blockIdx.x;       // 128 blocks
  const int c  = (int)threadIdx.x;      // 64 threads = one channel each
  const float* pp = partial + (size_t)bg * CHUNKS * PSTRIDE;

  float M = -3.0e38f;
  for (int k = 0; k < CHUNKS; ++k)
    M = fmaxf(M, pp[k * PSTRIDE + 64]);

  float S = 0.0f, acc = 0.0f;
  for (int k = 0; k < CHUNKS; ++k) {
    const float w = __expf(pp[k * PSTRIDE + 64] - M);
    S   += pp[k * PSTRIDE + 65] * w;
    acc += pp[k * PSTRIDE + c]  * w;
  }
  const int b = bg >> 3;
  const int g = bg & (G_ - 1);
  ctx[b * C_ + g * SHI + c] = acc / S;
}

// ------------------------------------------------------------------
// Kernel 3/5: out[16,512] = A[16,512] @ Wt[512,512] + bias using
// V_WMMA_F32_16X16X4_F32. One wave per 16-wide N tile; K-loop of
// 128 WMMAs. Layouts per CDNA5 ISA 7.12.2:
//   A 16x4 f32:  v0 = K0(l0-15)/K2(l16-31), v1 = K1/K3, M = lane%16
//   B 4x16 f32:  v0 = rowK0(l0-15)/rowK2(l16-31), v1 = K1/K3, N = lane%16
//   D 16x16 f32: vr = M=r (l0-15) / M=r+8 (l16-31), N = lane%16
// ------------------------------------------------------------------
__global__ void gemm16_wmma(const float* __restrict__ A,
                            const float* __restrict__ Wt,
                            const float* __restrict__ bias,
                            float* __restrict__ out) {
  const int n0   = (int)blockIdx.x * 16;
  const int lane = (int)(threadIdx.x & 31);
  const int half = lane >> 4;       // 0: K{0,1}, 1: K{2,3}
  const int l16  = lane & 15;

  v8f acc = {};
  for (int k = 0; k < C_; k += 4) {
    const int kk = k + 2 * half;
    v2f a = *(const v2f*)(A + (size_t)l16 * C_ + kk);  // A[M=l16][kk], [kk+1]
    v2f bf;
    bf.x = Wt[(size_t)(kk + 0) * C_ + n0 + l16];       // B[kk]  [N=l16]
    bf.y = Wt[(size_t)(kk + 1) * C_ + n0 + l16];       // B[kk+1][N=l16]
    acc = __builtin_amdgcn_wmma_f32_16x16x4_f32(
        /*neg_a=*/false, a, /*neg_b=*/false, bf,
        /*c_mod=*/(short)0, acc, /*reuse_a=*/false, /*reuse_b=*/false);
  }

  const float bv = bias[n0 + l16];
  #pragma unroll
  for (int r = 0; r < 8; ++r) {
    out[(size_t)(r + 8 * half) * C_ + n0 + l16] = acc[r] + bv;
  }
}

// ------------------------------------------------------------------
// Kernel 4: per-row LayerNorm(C=512) + ReLU
// ------------------------------------------------------------------
__global__ void ln_relu(const float* __restrict__ h,
                        const float* __restrict__ gamma,
                        const float* __restrict__ beta,
                        float* __restrict__ outv) {
  __shared__ float red[18];
  const int b   = (int)blockIdx.x;     // 16 rows
  const int tid = (int)threadIdx.x;    // 256 threads, 2 channels each
  const float v0 = h[b * C_ + tid];
  const float v1 = h[b * C_ + tid + 256];
  float sum = v0 + v1;
  float sq  = v0 * v0 + v1 * v1;
  for (int off = 16; off > 0; off >>= 1) {
    sum += __shfl_xor(sum, off, 32);
    sq  += __shfl_xor(sq,  off, 32);
  }
  const int wv = tid >> 5, ln = tid & 31;
  if (ln == 0) { red[wv] = sum; red[8 + wv] = sq; }
  __syncthreads();
  if (tid == 0) {
    float ts = 0.0f, tq = 0.0f;
    for (int i = 0; i < 8; ++i) { ts += red[i]; tq += red[8 + i]; }
    red[16] = ts; red[17] = tq;
  }
  __syncthreads();
  const float mu  = red[16] * (1.0f / C_);
  const float var = red[17] * (1.0f / C_) - mu * mu;
  const float inv = rsqrtf(var + LN_EPS);
  const float r0 = (v0 - mu) * inv * gamma[tid]       + beta[tid];
  const float r1 = (v1 - mu) * inv * gamma[tid + 256] + beta[tid + 256];
  outv[b * C_ + tid]       = fmaxf(r0, 0.0f);
  outv[b * C_ + tid + 256] = fmaxf(r1, 0.0f);
}

// ------------------------------------------------------------------
// Kernel 6: out = x + t[b, :] broadcast. Pure streaming: v4f vectors,
// non-temporal stores so the 252 MB output stream doesn't thrash L2.
// ------------------------------------------------------------------
__global__ void residual_add(const float* __restrict__ x,
                             const float* __restrict__ t,
                             float* __restrict__ out) {
  const size_t nvec   = (size_t)B_ * N_ * C_ / 4;
  const size_t stride = (size_t)gridDim.x * blockDim.x;
  for (size_t v = blockIdx.x * (size_t)blockDim.x + threadIdx.x; v < nvec; v += stride) {
    const size_t f = v * 4;
    const int c = (int)(f & (C_ - 1));
    const int b = (int)(f / ((size_t)N_ * C_));
    const v4f xv = *(const v4f*)(x + f);
    const v4f tv = *(const v4f*)(t + (size_t)b * C_ + c);
    const v4f o  = xv + tv;
    __builtin_nontemporal_store(o, (v4f*)(out + f));
  }
}

// ------------------------------------------------------------------
extern "C" void kernel_launch(void* const* d_in, const int* in_sizes, int n_in,
                              void* d_out, int out_size, void* d_ws, size_t ws_size,
                              hipStream_t stream) {
  const float* x      = (const float*)d_in[0];
  const float* w_mask = (const float*)d_in[1];
  const float* b_mask = (const float*)d_in[2];
  const float* w1     = (const float*)d_in[3];
  const float* b1     = (const float*)d_in[4];
  const float* gamma  = (const float*)d_in[5];
  const float* beta   = (const float*)d_in[6];
  const float* w2     = (const float*)d_in[7];
  const float* b2     = (const float*)d_in[8];
  float* out = (float*)d_out;

  // workspace layout (floats)
  float* partial = (float*)d_ws;                       // 8192 * 66
  float* ctx = partial + (size_t)B_ * G_ * CHUNKS * PSTRIDE;
  float* h   = ctx + B_ * C_;
  float* hn  = h   + B_ * C_;
  float* t   = hn  + B_ * C_;

  // 8192 chunk-waves, 8 waves per 256-thread block
  attn_pool_partial<<<(B_ * G_ * CHUNKS) / 8, 256, 0, stream>>>(x, w_mask, b_mask, partial);
  attn_combine<<<B_ * G_, SHI, 0, stream>>>(partial, ctx);
  gemm16_wmma<<<C_ / 16, 32, 0, stream>>>(ctx, w1, b1, h);
  ln_relu<<<B_, 256, 0, stream>>>(h, gamma, beta, hn);
  gemm16_wmma<<<C_ / 16, 32, 0, stream>>>(hn, w2, b2, t);
  residual_add<<<4096, 256, 0, stream>>>(x, t, out);
}